// ReverseHashModel_41351945126449
// MI455X (gfx1250) — compile-verified
//
#include <hip/hip_runtime.h>
#include <stdint.h>

// Problem constants (match reference: B, W, L, K = 256, 4096, 16, 8)
constexpr int Bx = 256;
constexpr int Wx = 4096;
constexpr int Lx = 16;
constexpr int Kx = 8;

constexpr int ROWS = 2;    // batch rows per workgroup
constexpr int NT   = 512;  // threads per block (16 wave32s)

// ---- CDNA5 async global<->LDS path (probe-confirmed present; params are v4i32*) ----
typedef int v4i __attribute__((ext_vector_type(4)));
typedef __attribute__((address_space(1))) v4i* gv4p;   // global
typedef __attribute__((address_space(3))) v4i* lv4p;   // LDS

#if __has_builtin(__builtin_amdgcn_global_load_async_to_lds_b128)
#define HAVE_ASYNC_LD 1
#endif
#if __has_builtin(__builtin_amdgcn_global_store_async_from_lds_b128)
#define HAVE_ASYNC_ST 1
#endif

#if __has_builtin(__builtin_amdgcn_s_wait_asynccnt)
#define WAIT_ASYNC0() __builtin_amdgcn_s_wait_asynccnt(0)
#else
#define WAIT_ASYNC0() asm volatile("s_wait_asynccnt 0" ::: "memory")
#endif

__global__ __launch_bounds__(NT) void rhm_kernel(
    const float* __restrict__ obs,       // (B, W)
    const float* __restrict__ prior_w,   // (W)
    const float* __restrict__ prior_b,   // (W)
    const float* __restrict__ weights,   // (L-1, W, K)
    const float* __restrict__ biases,    // (L-1, W)
    const int*   __restrict__ parents,   // (L-1, W, K)
    const unsigned char* __restrict__ inv_mask, // (L-1, W) bool (1 byte)
    float* __restrict__ out)             // (B, W)
{
    // Ping-pong h buffers: 2 * ROWS * 16KB = 64KB LDS (WGP has 320KB)
    __shared__ float hbuf[2][ROWS][Wx];

    const int t    = threadIdx.x;
    const int row0 = blockIdx.x * ROWS;

    // ---- Stage observed rows into hbuf[0] (async DMA global -> LDS) ----
    constexpr int VECS_PER_ROW = Wx / 4;            // float4 chunks per row
    constexpr int TOTAL_VECS   = ROWS * VECS_PER_ROW;
#if defined(HAVE_ASYNC_LD)
    for (int i = t; i < TOTAL_VECS; i += NT) {
        const int r = i / VECS_PER_ROW;
        const int c = i % VECS_PER_ROW;
        const float* g = obs + (size_t)(row0 + r) * Wx + (size_t)c * 4;
        __builtin_amdgcn_global_load_async_to_lds_b128(
            (gv4p)g, (lv4p)&hbuf[0][r][c * 4], 0, 0);
    }
    WAIT_ASYNC0();
#else
    for (int i = t; i < TOTAL_VECS; i += NT) {
        const int r = i / VECS_PER_ROW;
        const int c = i % VECS_PER_ROW;
        *(float4*)&hbuf[0][r][c * 4] =
            *(const float4*)(obs + (size_t)(row0 + r) * Wx + (size_t)c * 4);
    }
#endif
    __syncthreads();

    // ---- Level 0: h = relu(obs * prior_w + prior_b), hbuf[0] -> hbuf[1] ----
    for (int i = t; i < ROWS * Wx; i += NT) {
        const int r = i >> 12;          // Wx == 4096 == 2^12
        const int n = i & (Wx - 1);
        hbuf[1][r][n] = fmaxf(0.0f, fmaf(hbuf[0][r][n], prior_w[n], prior_b[n]));
    }
    __syncthreads();
    int cur = 1;

    // ---- 15 sparse-gather levels, fully LDS-resident ----
    for (int l = 0; l < Lx - 1; ++l) {
        const float*         wl = weights  + (size_t)l * Wx * Kx;
        const float*         bl = biases   + (size_t)l * Wx;
        const int*           pl = parents  + (size_t)l * Wx * Kx;
        const unsigned char* il = inv_mask + (size_t)l * Wx;
        const int nxt = cur ^ 1;

        // Prefetch next level's tables (128B cachelines; 4 nodes/line)
        if (l + 1 < Lx - 1) {
            const int*   pn = parents + (size_t)(l + 1) * Wx * Kx;
            const float* wn = weights + (size_t)(l + 1) * Wx * Kx;
            for (int i = t; i < Wx / 4; i += NT) {
                __builtin_prefetch(pn + (size_t)i * 32, 0, 0);
                __builtin_prefetch(wn + (size_t)i * 32, 0, 0);
            }
        }

        const float (*hprev)[Wx] = hbuf[cur];
        float       (*hnext)[Wx] = hbuf[nxt];

        for (int n = t; n < Wx; n += NT) {          // 8 nodes per thread
            const int4   p0 = ((const int4*)(pl + (size_t)n * Kx))[0];
            const int4   p1 = ((const int4*)(pl + (size_t)n * Kx))[1];
            const float4 w0 = ((const float4*)(wl + (size_t)n * Kx))[0];
            const float4 w1 = ((const float4*)(wl + (size_t)n * Kx))[1];
            const float  bias = bl[n];
            const bool   inv  = il[n] != 0;

            // where(inv, 1-x, x)·w == inv ? (Σw - Σ h·w) : Σ h·w
            const float wsum = (w0.x + w0.y) + (w0.z + w0.w)
                             + (w1.x + w1.y) + (w1.z + w1.w);
            const float c = inv ? (bias + wsum) : bias;
            const float s = inv ? -1.0f : 1.0f;

            #pragma unroll
            for (int r = 0; r < ROWS; ++r) {
                const float* hp = &hprev[r][0];
                float acc = hp[p0.x] * w0.x + hp[p0.y] * w0.y
                          + hp[p0.z] * w0.z + hp[p0.w] * w0.w
                          + hp[p1.x] * w1.x + hp[p1.y] * w1.y
                          + hp[p1.z] * w1.z + hp[p1.w] * w1.w;
                hnext[r][n] = fmaxf(0.0f, fmaf(s, acc, c));
            }
        }
        __syncthreads();
        cur = nxt;
    }

    // ---- Write final h rows out (async LDS -> global; drained before endpgm) ----
#if defined(HAVE_ASYNC_ST)
    for (int i = t; i < TOTAL_VECS; i += NT) {
        const int r = i / VECS_PER_ROW;
        const int c = i % VECS_PER_ROW;
        float* g = out + (size_t)(row0 + r) * Wx + (size_t)c * 4;
        __builtin_amdgcn_global_store_async_from_lds_b128(
            (gv4p)g, (lv4p)&hbuf[cur][r][c * 4], 0, 0);
    }
    WAIT_ASYNC0();
#else
    for (int i = t; i < TOTAL_VECS; i += NT) {
        const int r = i / VECS_PER_ROW;
        const int c = i % VECS_PER_ROW;
        *(float4*)(out + (size_t)(row0 + r) * Wx + (size_t)c * 4) =
            *(const float4*)&hbuf[cur][r][c * 4];
    }
#endif
}

extern "C" void kernel_launch(void* const* d_in, const int* in_sizes, int n_in,
                              void* d_out, int out_size, void* d_ws, size_t ws_size,
                              hipStream_t stream) {
    (void)in_sizes; (void)n_in; (void)out_size; (void)d_ws; (void)ws_size;

    const float*         obs = (const float*)d_in[0];         // observed_bits (B,W)
    const float*         pw  = (const float*)d_in[1];         // prior_w (W)
    const float*         pb  = (const float*)d_in[2];         // prior_b (W)
    const float*         wts = (const float*)d_in[3];         // weights (L-1,W,K)
    const float*         bia = (const float*)d_in[4];         // biases (L-1,W)
    const int*           par = (const int*)d_in[5];           // parents (L-1,W,K)
    const unsigned char* inv = (const unsigned char*)d_in[6]; // inv_mask (L-1,W) bool
    float*               out = (float*)d_out;                 // (B,W)

    dim3 grid(Bx / ROWS);
    dim3 block(NT);
    rhm_kernel<<<grid, block, 0, stream>>>(obs, pw, pb, wts, bia, par, inv, out);
}